// MyGCNConv_72086731096478
// MI455X (gfx1250) — compile-verified
//
#include <hip/hip_runtime.h>

typedef __attribute__((ext_vector_type(2))) float v2f;
typedef __attribute__((ext_vector_type(8))) float v8f;

#define C_IN  256
#define C_OUT 128

// ---------------- zero init (deg + out), grid-stride ----------------
__global__ void gcn_zero_kernel(float* __restrict__ deg, float* __restrict__ out,
                                int n_deg, long long n_out) {
  long long i = (long long)blockIdx.x * blockDim.x + threadIdx.x;
  long long stride = (long long)gridDim.x * blockDim.x;
  for (long long j = i; j < n_out; j += stride) out[j] = 0.0f;
  for (long long j = i; j < n_deg; j += stride) deg[j] = 0.0f;
}

// ---------------- degree via hardware f32 atomics ----------------
__global__ void gcn_degree_kernel(const long long* __restrict__ r,
                                  float* __restrict__ deg, int E) {
  int e = blockIdx.x * blockDim.x + threadIdx.x;
  if (e < E) unsafeAtomicAdd(&deg[(int)r[e]], 1.0f);
}

// ---------------- rsqrt(deg), 0 where deg==0 (in place) ----------------
__global__ void gcn_rsqrt_kernel(float* __restrict__ deg, int N) {
  int n = blockIdx.x * blockDim.x + threadIdx.x;
  if (n < N) {
    float d = deg[n];
    deg[n] = (d > 0.0f) ? rsqrtf(d) : 0.0f;
  }
}

// ---------------- h = x * W^T + b via V_WMMA_F32_16X16X4_F32 ----------------
// Block = 128 threads = 4 waves. Each block owns one 16-row M tile.
// Each wave owns two 16-col N tiles, reusing one A tile per two WMMAs.
__global__ void __launch_bounds__(128)
gcn_gemm_wmma_kernel(const float* __restrict__ x, const float* __restrict__ W,
                     const float* __restrict__ b, float* __restrict__ h, int N) {
  const int mtile = blockIdx.x;
  const int wave  = threadIdx.x >> 5;
  const int lane  = threadIdx.x & 31;
  const int m     = lane & 15;   // A row within tile / B col within tile
  const int khalf = lane >> 4;   // selects K pair {0,1} vs {2,3}

  const int ct0 = wave * 2;      // column tiles 0..7 across 4 waves
  const int ct1 = wave * 2 + 1;

  int rowA = mtile * 16 + m;
  if (rowA >= N) rowA = 0;       // clamp load; store handled below
  const int o0 = ct0 * 16 + m;   // W row supplying B-tile0 column m
  const int o1 = ct1 * 16 + m;

  const float* pa  = x + (size_t)rowA * C_IN + 2 * khalf;
  const float* pb0 = W + (size_t)o0   * C_IN + 2 * khalf;
  const float* pb1 = W + (size_t)o1   * C_IN + 2 * khalf;

  const float bias0 = b[o0];
  const float bias1 = b[o1];
  v8f c0 = {bias0, bias0, bias0, bias0, bias0, bias0, bias0, bias0};
  v8f c1 = {bias1, bias1, bias1, bias1, bias1, bias1, bias1, bias1};

#pragma unroll 4
  for (int k = 0; k < C_IN; k += 4) {
    v2f a  = *(const v2f*)(pa  + k);
    v2f b0 = *(const v2f*)(pb0 + k);
    v2f b1 = *(const v2f*)(pb1 + k);
    c0 = __builtin_amdgcn_wmma_f32_16x16x4_f32(false, a, false, b0,
                                               (short)0, c0, false, false);
    c1 = __builtin_amdgcn_wmma_f32_16x16x4_f32(false, a, false, b1,
                                               (short)0, c1, false, false);
  }

  // D layout: VGPR j -> row mtile*16 + 8*khalf + j, col = lane&15
  const int mBase = mtile * 16 + 8 * khalf;
  float* ph0 = h + (size_t)mBase * C_OUT + ct0 * 16 + m;
  float* ph1 = h + (size_t)mBase * C_OUT + ct1 * 16 + m;
  if (mtile * 16 + 16 <= N) {
    // full tile: block-uniform branch, straight-line stores w/ imm offsets
#pragma unroll
    for (int j = 0; j < 8; ++j) {
      ph0[(size_t)j * C_OUT] = c0[j];
      ph1[(size_t)j * C_OUT] = c1[j];
    }
  } else {
    // partial last tile (never taken when N % 16 == 0)
#pragma unroll
    for (int j = 0; j < 8; ++j) {
      if (mBase + j < N) {
        ph0[(size_t)j * C_OUT] = c0[j];
        ph1[(size_t)j * C_OUT] = c1[j];
      }
    }
  }
}

// ---------------- edge scatter: one wave32 per edge ----------------
// lane handles 4 channels: b128 gather of h[c], scale by ndr[c],
// 4x global_atomic_add_f32 into out[r]. h/out are L2-resident (192MB L2).
__global__ void gcn_scatter_kernel(const long long* __restrict__ r,
                                   const long long* __restrict__ cidx,
                                   const float* __restrict__ h,
                                   const float* __restrict__ ndr,
                                   float* __restrict__ out, int E) {
  const int lane   = threadIdx.x & 31;
  const int wave   = (blockIdx.x * blockDim.x + threadIdx.x) >> 5;
  const int nwaves = (gridDim.x * blockDim.x) >> 5;
  for (int e = wave; e < E; e += nwaves) {
    const int src = (int)cidx[e];
    const int dst = (int)r[e];
    const float s = ndr[src];
    const float4 msg = *(const float4*)(h + (size_t)src * C_OUT + lane * 4);
    float* o = out + (size_t)dst * C_OUT + lane * 4;
    unsafeAtomicAdd(o + 0, msg.x * s);
    unsafeAtomicAdd(o + 1, msg.y * s);
    unsafeAtomicAdd(o + 2, msg.z * s);
    unsafeAtomicAdd(o + 3, msg.w * s);
  }
}

// ---------------- final destination-side normalization ----------------
__global__ void gcn_scale_kernel(float* __restrict__ out,
                                 const float* __restrict__ ndr, int N) {
  int idx  = blockIdx.x * blockDim.x + threadIdx.x;
  int n    = idx >> 5;
  int lane = idx & 31;
  if (n < N) {
    float s = ndr[n];
    float4* p = (float4*)(out + (size_t)n * C_OUT + lane * 4);
    float4 v = *p;
    v.x *= s; v.y *= s; v.z *= s; v.w *= s;
    *p = v;
  }
}

extern "C" void kernel_launch(void* const* d_in, const int* in_sizes, int n_in,
                              void* d_out, int out_size, void* d_ws, size_t ws_size,
                              hipStream_t stream) {
  const float*     x  = (const float*)d_in[0];
  const float*     W  = (const float*)d_in[1];
  const float*     b  = (const float*)d_in[2];
  const long long* ei = (const long long*)d_in[3];   // int64 edge_index [2,E]
  float* out = (float*)d_out;

  const int N = in_sizes[0] / C_IN;   // 50000
  const int E = in_sizes[3] / 2;      // 625000
  const long long* r = ei;            // destination rows
  const long long* c = ei + E;        // source cols

  // workspace: deg/ndr [N floats] then h [N x C_OUT floats]
  float* deg = (float*)d_ws;
  size_t hoff = ((size_t)N * sizeof(float) + 255) & ~(size_t)255;
  float* h = (float*)((char*)d_ws + hoff);

  gcn_zero_kernel<<<2048, 256, 0, stream>>>(deg, out, N, (long long)N * C_OUT);
  gcn_degree_kernel<<<(E + 255) / 256, 256, 0, stream>>>(r, deg, E);
  gcn_rsqrt_kernel<<<(N + 255) / 256, 256, 0, stream>>>(deg, N);
  gcn_gemm_wmma_kernel<<<(N + 15) / 16, 128, 0, stream>>>(x, W, b, h, N);
  gcn_scatter_kernel<<<4096, 256, 0, stream>>>(r, c, h, deg, out, E);
  gcn_scale_kernel<<<(N * 32 + 255) / 256, 256, 0, stream>>>(out, deg, N);
}